// SOMVectorQuantizer_25194278159085
// MI455X (gfx1250) — compile-verified
//
#include <hip/hip_runtime.h>
#include <cmath>

typedef __attribute__((ext_vector_type(2))) float v2f;
typedef __attribute__((ext_vector_type(8))) float v8f;

#define GD       16          // grid side
#define NU       256         // G*G units
#define BATCH    32
#define DIM      12288
#define KCHUNKS  12
#define KCHUNK   (DIM / KCHUNKS)   // 1024

// ---------------------------------------------------------------------------
// Kernel 1: partial GEMM  dot[b,u] = x[b] . units[u]  via V_WMMA_F32_16X16X4_F32
// grid = (16 N-tiles, 12 K-chunks), block = 64 (2 waves; wave = one 16-row M tile)
// Also accumulates partial ||units[u]||^2 (wave 0 only).
// ---------------------------------------------------------------------------
__global__ __launch_bounds__(64) void som_gemm_partial(
    const float* __restrict__ X, const float* __restrict__ U,
    float* __restrict__ dotp, float* __restrict__ unormp)
{
    const int lane  = threadIdx.x & 31;
    const int mtile = threadIdx.x >> 5;     // 0 or 1 -> rows 0-15 / 16-31 of X
    const int ntile = blockIdx.x;           // 0..15  -> units 16*ntile ..
    const int chunk = blockIdx.y;           // 0..11  -> K range

    const int r    = lane & 15;             // row within tile
    const int hi   = lane >> 4;             // 0: K pair {0,1}, 1: K pair {2,3}
    const int koff = hi * 2;

    const float* xrow = X + (size_t)(mtile * 16 + r) * DIM;
    const float* urow = U + (size_t)(ntile * 16 + r) * DIM;
    const v2f* xp = (const v2f*)(xrow + koff);   // 8-byte aligned (koff even)
    const v2f* up = (const v2f*)(urow + koff);

    v8f c = {0.f, 0.f, 0.f, 0.f, 0.f, 0.f, 0.f, 0.f};
    float us = 0.f;

    const int s0 = (chunk * KCHUNK) >> 1;
    const int s1 = s0 + (KCHUNK >> 1);
#pragma unroll 4
    for (int s = s0; s < s1; s += 2) {      // k advances by 4 per WMMA
        v2f a = xp[s];
        v2f b = up[s];
        // D = A(16x4 f32) * B(4x16 f32) + C   -> v_wmma_f32_16x16x4_f32
        c = __builtin_amdgcn_wmma_f32_16x16x4_f32(
                /*neg_a=*/false, a, /*neg_b=*/false, b,
                /*c_mod=*/(short)0, c, /*reuse_a=*/false, /*reuse_b=*/false);
        if (mtile == 0) us = fmaf(b.x, b.x, fmaf(b.y, b.y, us));
    }

    // C layout: VGPR j -> (M = j + 8*hi, N = lane&15)
    float* out = dotp + (size_t)chunk * BATCH * NU;
#pragma unroll
    for (int j = 0; j < 8; ++j) {
        int m = mtile * 16 + hi * 8 + j;
        out[(size_t)m * NU + ntile * 16 + r] = c[j];
    }

    if (mtile == 0) {
        float other = __shfl_down(us, 16, 32);   // lane L gets lane L+16 (wave32)
        if (hi == 0)
            unormp[chunk * NU + ntile * 16 + r] = us + other;
    }
}

// ---------------------------------------------------------------------------
// Kernel 2: ||x[b]||^2, one block per batch element
// ---------------------------------------------------------------------------
__global__ __launch_bounds__(256) void som_xnorm(
    const float* __restrict__ X, float* __restrict__ xnorm)
{
    __shared__ float sm[256];
    const int b = blockIdx.x, t = threadIdx.x;
    const float* xr = X + (size_t)b * DIM;
    float s = 0.f;
    for (int k = t; k < DIM; k += 256) { float v = xr[k]; s = fmaf(v, v, s); }
    sm[t] = s; __syncthreads();
    for (int off = 128; off > 0; off >>= 1) {
        if (t < off) sm[t] += sm[t + off];
        __syncthreads();
    }
    if (t == 0) xnorm[b] = sm[0];
}

// ---------------------------------------------------------------------------
// Kernel 3: reduce partials -> feat_dist, argmin (first-index tie-break),
// gaussian neighborhood, scalar loss. Single block of 256 threads.
// ---------------------------------------------------------------------------
__global__ __launch_bounds__(256) void som_finalize(
    const float* __restrict__ dotp, const float* __restrict__ unormp,
    const float* __restrict__ xnorm,
    int* __restrict__ bmu_out, float* __restrict__ loss_out,
    float inv_two_sigma2, float temp2)
{
    __shared__ float feat[BATCH * NU];   // 32 KB
    __shared__ float un[NU];
    __shared__ float redv[256];
    __shared__ int   redi[256];
    const int t = threadIdx.x;

    float s = 0.f;
    for (int ch = 0; ch < KCHUNKS; ++ch) s += unormp[ch * NU + t];
    un[t] = s;
    __syncthreads();

    for (int idx = t; idx < BATCH * NU; idx += 256) {
        int b = idx >> 8, u = idx & 255;
        float d = 0.f;
        for (int ch = 0; ch < KCHUNKS; ++ch)
            d += dotp[((size_t)ch * BATCH + b) * NU + u];
        feat[idx] = xnorm[b] + un[u] - 2.f * d;
    }
    __syncthreads();

    float lossacc = 0.f;
    for (int b = 0; b < BATCH; ++b) {
        float v = feat[b * NU + t];
        redv[t] = v; redi[t] = t;
        __syncthreads();
        for (int off = 128; off > 0; off >>= 1) {
            if (t < off) {
                float v2 = redv[t + off]; int i2 = redi[t + off];
                if (v2 < redv[t] || (v2 == redv[t] && i2 < redi[t])) {
                    redv[t] = v2; redi[t] = i2;
                }
            }
            __syncthreads();
        }
        int bmu = redi[0];
        __syncthreads();               // everyone read redi[0] before reuse
        if (t == 0) bmu_out[b] = bmu;

        float dy = (float)(t >> 4) - (float)(bmu >> 4);
        float dx = (float)(t & 15) - (float)(bmu & 15);
        float pd = dy * dy + dx * dx;
        float g  = expf(-pd * inv_two_sigma2);
        if (g < 0.001f) g = 0.f;
        lossacc = fmaf(temp2 * g * g, v, lossacc);
    }

    redv[t] = lossacc;
    __syncthreads();
    for (int off = 128; off > 0; off >>= 1) {
        if (t < off) redv[t] += redv[t + off];
        __syncthreads();
    }
    if (t == 0)
        loss_out[0] = redv[0] * (1.0f / ((float)BATCH * (float)NU * (float)DIM));
}

// ---------------------------------------------------------------------------
extern "C" void kernel_launch(void* const* d_in, const int* in_sizes, int n_in,
                              void* d_out, int out_size, void* d_ws, size_t ws_size,
                              hipStream_t stream)
{
    (void)in_sizes; (void)n_in; (void)out_size; (void)ws_size;

    const float* X = (const float*)d_in[0];   // (32, 3, 64, 64) = (32, 12288)
    const float* U = (const float*)d_in[1];   // (16, 16, 12288) = (256, 12288)

    float* ws_dot = (float*)d_ws;                         // 12*32*256 floats
    float* ws_un  = ws_dot + (size_t)KCHUNKS * BATCH * NU; // 12*256 floats
    float* ws_xn  = ws_un + (size_t)KCHUNKS * NU;          // 32 floats

    int*   bmu_out  = (int*)d_out;            // first 32 slots: int32 bmu indices
    float* loss_out = (float*)d_out + BATCH;  // slot 32: float loss

    // Host-side constants (match the reference exactly)
    const double deno  = 1.0 / (0.9 * 10000.0 / log(16.0));
    const double ksize = 16.0 * exp(-2.0 * deno);
    const double sigma = 0.3 * ((ksize - 1.0) * 0.5 - 1.0) + 0.8;
    const double temp  = exp(-(2.0 * 2.0) / 10000.0);
    const float inv2s2 = (float)(1.0 / (2.0 * sigma * sigma));
    const float temp2  = (float)(temp * temp);

    som_gemm_partial<<<dim3(GD, KCHUNKS), 64, 0, stream>>>(X, U, ws_dot, ws_un);
    som_xnorm<<<BATCH, 256, 0, stream>>>(X, ws_xn);
    som_finalize<<<1, 256, 0, stream>>>(ws_dot, ws_un, ws_xn,
                                        bmu_out, loss_out, inv2s2, temp2);
}